// FlowMambaCell_54820962566306
// MI455X (gfx1250) — compile-verified
//
#include <hip/hip_runtime.h>
#include <math.h>

// ---------------------------------------------------------------------------
// FlowMambaCell for MI455X (gfx1250, wave32).
//   Pass 1: circular 3x3 convs (delta/B/C) as f32 WMMA GEMM (16x16x4).
//   Pass 2: bandwidth-bound SSM stream over s_prev -> (y_out, s_next).
// ---------------------------------------------------------------------------

typedef float v2f __attribute__((ext_vector_type(2)));
typedef float v8f __attribute__((ext_vector_type(8)));

#define D_MODEL 64
#define D_STATE 16
#define NPIX    1024      // 32*32
#define NUM_V   9
#define NBATCH  4

__device__ __forceinline__ float softplus_f(float x) {
    // jax.nn.softplus: log1p(exp(x)), stable for large x
    return (x > 20.0f) ? x : log1pf(__expf(x));
}

// ---------------------------------------------------------------------------
// Kernel 1: conv-as-GEMM with V_WMMA_F32_16X16X4_F32.
// Each wave handles one (batch, 16-pixel row segment) tile.
//   GEMM: D[o, m] = sum_k W[o, k] * U[k, m],  k = tap*64 + cin  (K = 576)
//   A-frag (weights, 16x4): lane(ln,half), vgpr j -> W[M=ln][K = c0+2*half+j]
//   B-frag (u, 4x16):       lane(ln,half), vgpr j -> U[K = c0+2*half+j][N=ln]
//   D (16x16): vgpr r -> row (r + 8*half), col ln
// ---------------------------------------------------------------------------
__global__ __launch_bounds__(128) void flowmamba_conv_wmma(
    const float* __restrict__ u,       // (4,64,32,32)
    const float* __restrict__ wd,      // (64,64,3,3)
    const float* __restrict__ bd,      // (64,)
    const float* __restrict__ wB,      // (16,64,3,3)
    const float* __restrict__ wC,      // (16,64,3,3)
    const float* __restrict__ dtp,     // scalar
    float* __restrict__ wsDelta,       // (4,64,32,32) softplus(delta)
    float* __restrict__ wsB,           // (4,16,32,32)
    float* __restrict__ wsC)           // (4,16,32,32)
{
    const int lane = threadIdx.x & 31;
    const int wave = threadIdx.x >> 5;
    const int tile = blockIdx.x * 4 + wave;   // 0..255
    const int b    = tile >> 6;
    const int rem  = tile & 63;
    const int y    = rem >> 1;                // row 0..31
    const int x0   = (rem & 1) << 4;          // 0 or 16
    const int half = lane >> 4;               // 0 or 1
    const int ln   = lane & 15;

    // circular-wrap coordinates for the 3 taps in each direction
    int xs[3], ys[3];
#pragma unroll
    for (int d = 0; d < 3; ++d) {
        xs[d] = (x0 + ln + d - 1) & 31;       // column for pixel m=ln, dx=d-1
        ys[d] = (y + d - 1) & 31;             // row for dy=d-1
    }

    const float dtinv = dtp[0];
    const float* ub = u + b * (D_MODEL * NPIX);

    v8f accD[4] = {};     // delta: 4 tiles of 16 output channels
    v8f accB = {};        // B_val: 16 output channels
    v8f accC = {};        // C_val: 16 output channels

    for (int t = 0; t < 9; ++t) {
        const int ky = t / 3, kx = t - 3 * ky;
        const float* urow = ub + ys[ky] * 32;
        const int ux = xs[kx];
        for (int c0 = 0; c0 < 64; c0 += 4) {
            const int cu = c0 + half * 2;     // this lane's K channels
            v2f uf;
            uf[0] = urow[(cu + 0) * NPIX + ux];
            uf[1] = urow[(cu + 1) * NPIX + ux];

            // 4 delta output-channel tiles share the u fragment
#pragma unroll
            for (int ot = 0; ot < 4; ++ot) {
                const float* wp = wd + (ot * 16 + ln) * 576 + cu * 9 + t;
                v2f wf; wf[0] = wp[0]; wf[1] = wp[9];   // cin stride = 9
                accD[ot] = __builtin_amdgcn_wmma_f32_16x16x4_f32(
                    false, wf, false, uf, (short)0, accD[ot], false, false);
            }
            {
                const float* wp = wB + ln * 576 + cu * 9 + t;
                v2f wf; wf[0] = wp[0]; wf[1] = wp[9];
                accB = __builtin_amdgcn_wmma_f32_16x16x4_f32(
                    false, wf, false, uf, (short)0, accB, false, false);
            }
            {
                const float* wp = wC + ln * 576 + cu * 9 + t;
                v2f wf; wf[0] = wp[0]; wf[1] = wp[9];
                accC = __builtin_amdgcn_wmma_f32_16x16x4_f32(
                    false, wf, false, uf, (short)0, accC, false, false);
            }
        }
    }

    // Epilogue: D vgpr r -> output channel (base + 8*half + r), pixel = ln.
    const int pix = y * 32 + x0 + ln;
#pragma unroll
    for (int ot = 0; ot < 4; ++ot) {
#pragma unroll
        for (int r = 0; r < 8; ++r) {
            const int o = ot * 16 + half * 8 + r;
            const float val = accD[ot][r] + bd[o] + dtinv;
            wsDelta[(b * D_MODEL + o) * NPIX + pix] = softplus_f(val);
        }
    }
#pragma unroll
    for (int r = 0; r < 8; ++r) {
        const int o = half * 8 + r;
        wsB[(b * D_STATE + o) * NPIX + pix] = accB[r];
        wsC[(b * D_STATE + o) * NPIX + pix] = accC[r];
    }
}

// ---------------------------------------------------------------------------
// Kernel 2: single-pass SSM stream. One thread per (b, c, pixel).
// A_bar/B_bar/C computed once in registers, reused across the 9 offsets.
// s_prev/s_next touched exactly once each, coalesced along x.
// ---------------------------------------------------------------------------
__global__ __launch_bounds__(256) void flowmamba_ssm_stream(
    const float* __restrict__ u,       // (4,64,32,32)
    const float* __restrict__ s_prev,  // (4,9,64,16,32,32)
    const float* __restrict__ logA,    // (64,16)
    const float* __restrict__ Dp,      // (64,)
    const float* __restrict__ wsDelta,
    const float* __restrict__ wsB,
    const float* __restrict__ wsC,
    float* __restrict__ y_out,         // (4,9,64,32,32)
    float* __restrict__ s_next)        // (4,9,64,16,32,32)
{
    const int blk = blockIdx.x;        // (b*64 + c)*4 + pb
    const int pb  = blk & 3;
    const int bc  = blk >> 2;
    const int c   = bc & 63;
    const int b   = bc >> 6;
    const int pix = pb * 256 + threadIdx.x;
    const int py  = pix >> 5;
    const int px  = pix & 31;

    const float delta = wsDelta[(b * D_MODEL + c) * NPIX + pix];
    const float uval  = u[(b * D_MODEL + c) * NPIX + pix];
    const float yskip = uval * Dp[c];

    float abar[D_STATE], bbar_u[D_STATE], cv[D_STATE];
#pragma unroll
    for (int n = 0; n < D_STATE; ++n) {
        const float A  = -__expf(logA[c * D_STATE + n]);   // strictly negative
        const float ab = __expf(delta * A);
        abar[n]   = ab;
        bbar_u[n] = (ab - 1.0f) / A * wsB[(b * D_STATE + n) * NPIX + pix] * uval;
        cv[n]     = wsC[(b * D_STATE + n) * NPIX + pix];
    }

    for (int v = 0; v < NUM_V; ++v) {
        const int vx = v / 3 - 1;                  // V_LIST order: x outer, y inner
        const int vy = v - 3 * (v / 3) - 1;
        // roll(s, (-vy,-vx)) : dest (y,x) <- src ((y+vy)%32, (x+vx)%32)
        const int sy = (py + vy) & 31;
        const int sx = (px + vx) & 31;
        const int sp = sy * 32 + sx;
        const int sbase = (((b * NUM_V + v) * D_MODEL + c) * D_STATE) * NPIX;
        float acc = 0.0f;
#pragma unroll
        for (int n = 0; n < D_STATE; ++n) {
            const float s  = s_prev[sbase + n * NPIX + sp];
            const float sn = fmaf(abar[n], s, bbar_u[n]);
            acc = fmaf(sn, cv[n], acc);
            s_next[sbase + n * NPIX + pix] = sn;
        }
        y_out[((b * NUM_V + v) * D_MODEL + c) * NPIX + pix] = acc + yskip;
    }
}

// ---------------------------------------------------------------------------
extern "C" void kernel_launch(void* const* d_in, const int* in_sizes, int n_in,
                              void* d_out, int out_size, void* d_ws, size_t ws_size,
                              hipStream_t stream) {
    const float* u_t  = (const float*)d_in[0];
    const float* s_pr = (const float*)d_in[1];
    const float* wd   = (const float*)d_in[2];
    const float* bd   = (const float*)d_in[3];
    const float* wB   = (const float*)d_in[4];
    const float* wC   = (const float*)d_in[5];
    const float* lA   = (const float*)d_in[6];
    const float* Dp   = (const float*)d_in[7];
    const float* dt   = (const float*)d_in[8];

    float* wsDelta = (float*)d_ws;                          // 4*64*1024 floats
    float* wsB     = wsDelta + NBATCH * D_MODEL * NPIX;     // 4*16*1024
    float* wsC     = wsB + NBATCH * D_STATE * NPIX;         // 4*16*1024

    float* y_out  = (float*)d_out;                          // 4*9*64*1024
    float* s_next = y_out + NBATCH * NUM_V * D_MODEL * NPIX;

    // Pass 1: 256 tiles (4 batches x 64 half-rows), 4 waves/block
    flowmamba_conv_wmma<<<64, 128, 0, stream>>>(
        u_t, wd, bd, wB, wC, dt, wsDelta, wsB, wsC);

    // Pass 2: 4*64 (b,c) planes x 4 pixel-blocks, 256 threads each
    flowmamba_ssm_stream<<<NBATCH * D_MODEL * 4, 256, 0, stream>>>(
        u_t, s_pr, lA, Dp, wsDelta, wsB, wsC, y_out, s_next);
}